// SelfAttention_2688649527299
// MI455X (gfx1250) — compile-verified
//
#include <hip/hip_runtime.h>
#include <hip/hip_bf16.h>

typedef float v2f __attribute__((ext_vector_type(2)));
typedef float v8f __attribute__((ext_vector_type(8)));

static __device__ __forceinline__ v8f wmma4(v2f a, v2f b, v8f c) {
  // D = A(16x4,f32) * B(4x16,f32) + C(16x16,f32)
  return __builtin_amdgcn_wmma_f32_16x16x4_f32(false, a, false, b, (short)0, c,
                                               false, false);
}

#define LD 512           // all row-major matrices here have 512 columns
#define ROWS 4096        // b*t
#define CHUNK 64
#define NCHUNK 32        // 2048 / 64
#define NTASK 512        // b*h*NCHUNK = 2*8*32

// ---------------------------------------------------------------------------
// Generic f32 GEMM: C[M,N] = A[M,K] @ B[K,N] (+bias), 64x64 tile per 128-thr WG
// ---------------------------------------------------------------------------
__global__ __launch_bounds__(128) void gemm_f32(const float* __restrict__ A,
                                                const float* __restrict__ B,
                                                float* __restrict__ C,
                                                const float* __restrict__ bias,
                                                int M, int N, int K) {
  __shared__ float As[64][17];
  __shared__ float Bs[16][65];
  const int t = threadIdx.x;
  const int wave = t >> 5, lane = t & 31, lh = lane >> 4, l16 = lane & 15;
  const int m0 = blockIdx.y * 64, n0 = blockIdx.x * 64;

  v8f acc[4] = {};

  for (int k0 = 0; k0 < K; k0 += 16) {
    {  // A tile 64x16: thread -> half row
      int r = t >> 1, c = (t & 1) * 8;
      const float4* src = reinterpret_cast<const float4*>(&A[(size_t)(m0 + r) * K + k0 + c]);
      if (k0 + 16 < K) __builtin_prefetch(&A[(size_t)(m0 + r) * K + k0 + 16 + c], 0, 1);
      float4 v0 = src[0], v1 = src[1];
      float* d = &As[r][c];
      d[0] = v0.x; d[1] = v0.y; d[2] = v0.z; d[3] = v0.w;
      d[4] = v1.x; d[5] = v1.y; d[6] = v1.z; d[7] = v1.w;
    }
    {  // B tile 16x64
      int r = t >> 3, c = (t & 7) * 8;
      const float4* src = reinterpret_cast<const float4*>(&B[(size_t)(k0 + r) * N + n0 + c]);
      float4 v0 = src[0], v1 = src[1];
      float* d = &Bs[r][c];
      d[0] = v0.x; d[1] = v0.y; d[2] = v0.z; d[3] = v0.w;
      d[4] = v1.x; d[5] = v1.y; d[6] = v1.z; d[7] = v1.w;
    }
    __syncthreads();
    const int mr = wave * 16 + l16;
#pragma unroll
    for (int kk = 0; kk < 16; kk += 4) {
      v2f a;
      a.x = As[mr][kk + 2 * lh];
      a.y = As[mr][kk + 2 * lh + 1];
#pragma unroll
      for (int nt = 0; nt < 4; ++nt) {
        v2f b;
        b.x = Bs[kk + 2 * lh][nt * 16 + l16];
        b.y = Bs[kk + 2 * lh + 1][nt * 16 + l16];
        acc[nt] = wmma4(a, b, acc[nt]);
      }
    }
    __syncthreads();
  }

#pragma unroll
  for (int nt = 0; nt < 4; ++nt) {
    int col = n0 + nt * 16 + l16;
    float badd = (bias != nullptr) ? bias[col] : 0.0f;
#pragma unroll
    for (int r = 0; r < 8; ++r) {
      int row = m0 + wave * 16 + r + 8 * lh;
      C[(size_t)row * N + col] = acc[nt][r] + badd;
    }
  }
}

// ---------------------------------------------------------------------------
// q <- softmax(q, per 64-d head) * 8^-1 ; k <- exp(k). One head-row per wave.
// ---------------------------------------------------------------------------
__global__ __launch_bounds__(128) void qk_transform(float* __restrict__ Q,
                                                    float* __restrict__ Kb) {
  const int wave = threadIdx.x >> 5, lane = threadIdx.x & 31;
  const int hr = blockIdx.x * 4 + wave;   // 0 .. 32767
  if (hr >= ROWS * 8) return;
  const int row = hr >> 3, h = hr & 7;
  const size_t base = (size_t)row * LD + h * 64;

  float q0 = Q[base + lane], q1 = Q[base + lane + 32];
  float m = fmaxf(q0, q1);
#pragma unroll
  for (int off = 16; off > 0; off >>= 1) m = fmaxf(m, __shfl_xor(m, off, 32));
  float e0 = __expf(q0 - m), e1 = __expf(q1 - m);
  float s = e0 + e1;
#pragma unroll
  for (int off = 16; off > 0; off >>= 1) s += __shfl_xor(s, off, 32);
  float inv = 0.125f / s;    // * dh^-0.5 = 1/8
  Q[base + lane] = e0 * inv;
  Q[base + lane + 32] = e1 * inv;

  float k0 = Kb[base + lane], k1 = Kb[base + lane + 32];
  Kb[base + lane] = __expf(k0);
  Kb[base + lane + 32] = __expf(k1);
}

// ---------------------------------------------------------------------------
// Pass A: per (b,h,chunk) local state Sloc = Kc^T @ Vc (64x64), ksum (64)
// ---------------------------------------------------------------------------
__global__ __launch_bounds__(256) void chunk_state(const float* __restrict__ Kb,
                                                   const float* __restrict__ Vb,
                                                   float* __restrict__ Sbuf,
                                                   float* __restrict__ ksumbuf) {
  __shared__ float Ks[64][65];
  __shared__ float Vs[64][65];
  const int task = blockIdx.x;
  const int ck = task & 31, h = (task >> 5) & 7, b = task >> 8;
  const int row0 = b * 2048 + ck * CHUNK, col0 = h * 64;
  const int t = threadIdx.x, wave = t >> 5, lane = t & 31;
  const int lh = lane >> 4, l16 = lane & 15;

  {  // load K,V chunk tiles (64x64 each)
    int r = t >> 2, c = (t & 3) * 16;
    const float4* ks = reinterpret_cast<const float4*>(&Kb[(size_t)(row0 + r) * LD + col0 + c]);
    const float4* vs = reinterpret_cast<const float4*>(&Vb[(size_t)(row0 + r) * LD + col0 + c]);
#pragma unroll
    for (int i = 0; i < 4; ++i) {
      float4 kv = ks[i], vv = vs[i];
      float* kd = &Ks[r][c + i * 4];
      kd[0] = kv.x; kd[1] = kv.y; kd[2] = kv.z; kd[3] = kv.w;
      float* vd = &Vs[r][c + i * 4];
      vd[0] = vv.x; vd[1] = vv.y; vd[2] = vv.z; vd[3] = vv.w;
    }
  }
  __syncthreads();

  const size_t sbase = (size_t)task * 4096;
#pragma unroll
  for (int tt = wave * 2; tt <= wave * 2 + 1; ++tt) {
    const int mt = tt >> 2, nt = tt & 3;
    v8f acc = {};
#pragma unroll
    for (int kk = 0; kk < 64; kk += 4) {
      int tr = kk + 2 * lh;
      v2f a, bfr;
      a.x = Ks[tr][mt * 16 + l16];      // A = Kc^T : A[d][t] = Kc[t][d]
      a.y = Ks[tr + 1][mt * 16 + l16];
      bfr.x = Vs[tr][nt * 16 + l16];    // B = Vc
      bfr.y = Vs[tr + 1][nt * 16 + l16];
      acc = wmma4(a, bfr, acc);
    }
#pragma unroll
    for (int r = 0; r < 8; ++r) {
      int d = mt * 16 + r + 8 * lh, n = nt * 16 + l16;
      Sbuf[sbase + d * 64 + n] = acc[r];
    }
  }
  if (t < 64) {
    float s = 0.f;
#pragma unroll 8
    for (int j = 0; j < 64; ++j) s += Ks[j][t];
    ksumbuf[task * 64 + t] = s;
  }
}

// ---------------------------------------------------------------------------
// Pass B: in-place exclusive prefix over the 32 chunks of every (b,h)
// ---------------------------------------------------------------------------
__global__ void prefix_scan(float* __restrict__ Sbuf, float* __restrict__ ksumbuf) {
  const int g = blockIdx.x * blockDim.x + threadIdx.x;
  if (g < 16 * 4096) {
    const int bh = g >> 12, e = g & 4095;
    float run = 0.f;
    for (int c = 0; c < NCHUNK; ++c) {
      size_t idx = (size_t)(bh * NCHUNK + c) * 4096 + e;
      float v = Sbuf[idx];
      Sbuf[idx] = run;
      run += v;
    }
  } else if (g < 16 * 4096 + 16 * 64) {
    const int s2 = g - 16 * 4096, bh = s2 >> 6, d = s2 & 63;
    float run = 0.f;
    for (int c = 0; c < NCHUNK; ++c) {
      int idx = (bh * NCHUNK + c) * 64 + d;
      float v = ksumbuf[idx];
      ksumbuf[idx] = run;
      run += v;
    }
  }
}

// ---------------------------------------------------------------------------
// Pass C: attn = (mask(Qc@Kc^T)@Vc + Qc@S_excl) / max(den, 1e-3)
// ---------------------------------------------------------------------------
__global__ __launch_bounds__(256) void attn_chunk(const float* __restrict__ Qb,
                                                  const float* __restrict__ Kb,
                                                  const float* __restrict__ Vb,
                                                  const float* __restrict__ Sbuf,
                                                  const float* __restrict__ ksumbuf,
                                                  float* __restrict__ attn) {
  extern __shared__ float sm[];
  float* Qs   = sm;                 // [64][65]
  float* Ks   = Qs + 64 * 65;
  float* Vs   = Ks + 64 * 65;
  float* Ss   = Vs + 64 * 65;
  float* Ms   = Ss + 64 * 65;
  float* ksum = Ms + 64 * 65;       // [64]
  float* Dinv = ksum + 64;          // [64]

  const int task = blockIdx.x;
  const int ck = task & 31, h = (task >> 5) & 7, b = task >> 8;
  const int row0 = b * 2048 + ck * CHUNK, col0 = h * 64;
  const int t = threadIdx.x, wave = t >> 5, lane = t & 31;
  const int lh = lane >> 4, l16 = lane & 15;

  {  // load Q,K,V chunk tiles + exclusive state + ksum
    int r = t >> 2, c = (t & 3) * 16;
    const float4* qs = reinterpret_cast<const float4*>(&Qb[(size_t)(row0 + r) * LD + col0 + c]);
    const float4* ks = reinterpret_cast<const float4*>(&Kb[(size_t)(row0 + r) * LD + col0 + c]);
    const float4* vs = reinterpret_cast<const float4*>(&Vb[(size_t)(row0 + r) * LD + col0 + c]);
    const float4* ss = reinterpret_cast<const float4*>(&Sbuf[(size_t)task * 4096 + t * 16]);
#pragma unroll
    for (int i = 0; i < 4; ++i) {
      float4 qv = qs[i], kv = ks[i], vv = vs[i], sv = ss[i];
      float* qd = &Qs[r * 65 + c + i * 4];
      qd[0] = qv.x; qd[1] = qv.y; qd[2] = qv.z; qd[3] = qv.w;
      float* kd = &Ks[r * 65 + c + i * 4];
      kd[0] = kv.x; kd[1] = kv.y; kd[2] = kv.z; kd[3] = kv.w;
      float* vd = &Vs[r * 65 + c + i * 4];
      vd[0] = vv.x; vd[1] = vv.y; vd[2] = vv.z; vd[3] = vv.w;
      float* sd = &Ss[(t >> 2) * 65 + (t & 3) * 16 + i * 4];
      sd[0] = sv.x; sd[1] = sv.y; sd[2] = sv.z; sd[3] = sv.w;
    }
    if (t < 64) ksum[t] = ksumbuf[task * 64 + t];
  }
  __syncthreads();

  // ---- Afull = Qc @ Kc^T, masked (j <= i) into Ms ----
#pragma unroll
  for (int tt = wave * 2; tt <= wave * 2 + 1; ++tt) {
    const int mt = tt >> 2, nt = tt & 3;
    v8f acc = {};
#pragma unroll
    for (int kk = 0; kk < 64; kk += 4) {
      v2f a, bfr;
      a.x = Qs[(mt * 16 + l16) * 65 + kk + 2 * lh];
      a.y = Qs[(mt * 16 + l16) * 65 + kk + 2 * lh + 1];
      bfr.x = Ks[(nt * 16 + l16) * 65 + kk + 2 * lh];   // B = Kc^T
      bfr.y = Ks[(nt * 16 + l16) * 65 + kk + 2 * lh + 1];
      acc = wmma4(a, bfr, acc);
    }
#pragma unroll
    for (int r = 0; r < 8; ++r) {
      int i = mt * 16 + r + 8 * lh, j = nt * 16 + l16;
      Ms[i * 65 + j] = (j <= i) ? acc[r] : 0.0f;
    }
  }
  __syncthreads();

  // ---- denominator ----
  if (t < 64) {
    float s = 0.f, dq = 0.f;
#pragma unroll 8
    for (int j = 0; j < 64; ++j) s += Ms[t * 65 + j];
#pragma unroll 8
    for (int d = 0; d < 64; ++d) dq += Qs[t * 65 + d] * ksum[d];
    Dinv[t] = 1.0f / fmaxf(s + dq, 1e-3f);
  }
  __syncthreads();

  // ---- numerator = Ms @ Vc + Qc @ S_excl ; scale and store ----
#pragma unroll
  for (int tt = wave * 2; tt <= wave * 2 + 1; ++tt) {
    const int mt = tt >> 2, nt = tt & 3;
    v8f acc = {};
#pragma unroll
    for (int kk = 0; kk < 64; kk += 4) {
      v2f a, bfr;
      a.x = Ms[(mt * 16 + l16) * 65 + kk + 2 * lh];
      a.y = Ms[(mt * 16 + l16) * 65 + kk + 2 * lh + 1];
      bfr.x = Vs[(kk + 2 * lh) * 65 + nt * 16 + l16];
      bfr.y = Vs[(kk + 2 * lh + 1) * 65 + nt * 16 + l16];
      acc = wmma4(a, bfr, acc);
    }
#pragma unroll
    for (int kk = 0; kk < 64; kk += 4) {
      v2f a, bfr;
      a.x = Qs[(mt * 16 + l16) * 65 + kk + 2 * lh];
      a.y = Qs[(mt * 16 + l16) * 65 + kk + 2 * lh + 1];
      bfr.x = Ss[(kk + 2 * lh) * 65 + nt * 16 + l16];
      bfr.y = Ss[(kk + 2 * lh + 1) * 65 + nt * 16 + l16];
      acc = wmma4(a, bfr, acc);
    }
#pragma unroll
    for (int r = 0; r < 8; ++r) {
      int i = mt * 16 + r + 8 * lh;
      int col = col0 + nt * 16 + l16;
      attn[(size_t)(row0 + i) * LD + col] = acc[r] * Dinv[i];
    }
  }
}

// ---------------------------------------------------------------------------
extern "C" void kernel_launch(void* const* d_in, const int* in_sizes, int n_in,
                              void* d_out, int out_size, void* d_ws, size_t ws_size,
                              hipStream_t stream) {
  const float* x  = (const float*)d_in[0];
  const float* Wq = (const float*)d_in[1];
  const float* Wk = (const float*)d_in[2];
  const float* Wv = (const float*)d_in[3];
  const float* Wo = (const float*)d_in[4];
  const float* bo = (const float*)d_in[5];
  float* out = (float*)d_out;

  float* ws = (float*)d_ws;
  const size_t MN = (size_t)ROWS * LD;   // 2,097,152 floats
  float* Qb = ws;
  float* Kb = Qb + MN;
  float* Vb = Kb + MN;
  float* At = Vb + MN;
  float* Sb = At + MN;                   // NTASK * 4096 floats
  float* Kc = Sb + (size_t)NTASK * 4096; // NTASK * 64 floats

  dim3 ggrid(LD / 64, ROWS / 64);        // (8, 64)

  gemm_f32<<<ggrid, 128, 0, stream>>>(x, Wq, Qb, nullptr, ROWS, LD, LD);
  gemm_f32<<<ggrid, 128, 0, stream>>>(x, Wk, Kb, nullptr, ROWS, LD, LD);
  gemm_f32<<<ggrid, 128, 0, stream>>>(x, Wv, Vb, nullptr, ROWS, LD, LD);

  qk_transform<<<(ROWS * 8) / 4, 128, 0, stream>>>(Qb, Kb);

  chunk_state<<<NTASK, 256, 0, stream>>>(Kb, Vb, Sb, Kc);

  prefix_scan<<<(16 * 4096 + 16 * 64 + 255) / 256, 256, 0, stream>>>(Sb, Kc);

  size_t shbytes = (size_t)(5 * 64 * 65 + 128) * sizeof(float);  // ~83.7 KB
  attn_chunk<<<NTASK, 256, shbytes, stream>>>(Qb, Kb, Vb, Sb, Kc, At);

  gemm_f32<<<ggrid, 128, 0, stream>>>(At, Wo, out, bo, ROWS, LD, LD);
}